// EdgeConvQuaternionMergeMotion_76836964926175
// MI455X (gfx1250) — compile-verified
//
#include <hip/hip_runtime.h>
#include <hip/hip_bf16.h>
#include <math.h>

typedef __attribute__((ext_vector_type(2))) float v2f;
typedef __attribute__((ext_vector_type(8))) float v8f;

#define NPTS  4096
#define NB    8
#define KNN   20
#define PANEL 512
#define NEG_INF (-__builtin_huge_valf())

// ---------------------------------------------------------------------------
// Kernel 0: squared norms xx[b][n] = |p|^2
// ---------------------------------------------------------------------------
__global__ __launch_bounds__(256) void xx_kernel(const float* __restrict__ pts,
                                                 float* __restrict__ xx) {
  int t = blockIdx.x * 256 + threadIdx.x;
  if (t < NB * NPTS) {
    const float* p = pts + t * 4;
    xx[t] = p[0] * p[0] + p[1] * p[1] + p[2] * p[2] + p[3] * p[3];
  }
}

// ---------------------------------------------------------------------------
// Kernel 1: kNN via WMMA f32 16x16x4 distance tiles + per-row top-20
// Block = 128 threads (4 waves) handles 16 rows of one batch.
// ---------------------------------------------------------------------------
__global__ __launch_bounds__(128) void knn_kernel(const float* __restrict__ pts,
                                                  const float* __restrict__ xx,
                                                  int* __restrict__ knn_idx) {
  __shared__ float s_pd[16 * PANEL];      // 32 KB distance panel
  __shared__ float s_xxi[16];
  __shared__ float s_xxj[PANEL];
  __shared__ float s_mv[16 * 8 * KNN];    // 10 KB merge values
  __shared__ int   s_mi[16 * 8 * KNN];    // 10 KB merge indices

  const int tid  = threadIdx.x;
  const int lane = tid & 31;
  const int wave = tid >> 5;
  const int b    = blockIdx.x >> 8;        // 256 row-tiles per batch
  const int i0   = (blockIdx.x & 255) << 4;
  const int base = b * NPTS;

  if (tid < 16) s_xxi[tid] = xx[base + i0 + tid];

  // A fragment: 16x4 f32 rows i0..i0+15  (lane<16 -> K=0,1 ; lane>=16 -> K=2,3)
  const int mrow  = lane & 15;
  const int kbase = (lane < 16) ? 0 : 2;
  v2f afrag;
  afrag.x = pts[(base + i0 + mrow) * 4 + kbase];
  afrag.y = pts[(base + i0 + mrow) * 4 + kbase + 1];

  // per-thread running top-20 (row r, segment s of 8)
  const int r = tid >> 3;
  const int s = tid & 7;
  float vals[KNN];
  int   idxs[KNN];
#pragma unroll
  for (int q = 0; q < KNN; ++q) { vals[q] = NEG_INF; idxs[q] = 0; }

  for (int p = 0; p < NPTS; p += PANEL) {
    for (int c = tid; c < PANEL; c += 128) s_xxj[c] = xx[base + p + c];
    __syncthreads();

    // 32 column tiles per panel, 8 per wave
    for (int t = wave * 8; t < wave * 8 + 8; ++t) {
      const int j0 = p + t * 16;
      v2f bfrag;   // B 4x16: lanes<16 rows K=0,1 ; lanes>=16 rows K=2,3
      bfrag.x = pts[(base + j0 + mrow) * 4 + kbase];
      bfrag.y = pts[(base + j0 + mrow) * 4 + kbase + 1];
      v8f acc = {0.f, 0.f, 0.f, 0.f, 0.f, 0.f, 0.f, 0.f};
      acc = __builtin_amdgcn_wmma_f32_16x16x4_f32(false, afrag, false, bfrag,
                                                  (short)0, acc, false, false);
      const int colL = t * 16 + mrow;       // panel-local column
      const int hi   = (lane >> 4) << 3;    // 0 or 8
      const float xxj = s_xxj[colL];
#pragma unroll
      for (int v = 0; v < 8; ++v) {
        const int row = v + hi;
        s_pd[row * PANEL + colL] = 2.0f * acc[v] - s_xxi[row] - xxj;
      }
    }
    __syncthreads();

    // top-K update: 8 threads/row, 64 candidates each
    const int cbase = s * 64;
#pragma unroll 1
    for (int c = 0; c < 64; ++c) {
      const int colL = cbase + c;
      float v = s_pd[r * PANEL + colL];
      if (v > vals[KNN - 1]) {
        vals[KNN - 1] = v;
        idxs[KNN - 1] = p + colL;
#pragma unroll
        for (int q = KNN - 1; q > 0; --q) {
          if (vals[q] > vals[q - 1]) {
            float tv = vals[q]; vals[q] = vals[q - 1]; vals[q - 1] = tv;
            int   ti = idxs[q]; idxs[q] = idxs[q - 1]; idxs[q - 1] = ti;
          }
        }
      }
    }
    __syncthreads();
  }

  // publish per-thread sorted lists
#pragma unroll
  for (int q = 0; q < KNN; ++q) {
    s_mv[(r * 8 + s) * KNN + q] = vals[q];
    s_mi[(r * 8 + s) * KNN + q] = idxs[q];
  }
  __syncthreads();

  // 8-way merge of sorted lists -> global top-20 per row
  if (tid < 16) {
    const int rr = tid;
    int p8[8];
#pragma unroll
    for (int q = 0; q < 8; ++q) p8[q] = 0;
    int* outp = knn_idx + (base + i0 + rr) * KNN;
    for (int sel = 0; sel < KNN; ++sel) {
      float best = NEG_INF;
      int bs = 0;
#pragma unroll
      for (int q = 0; q < 8; ++q) {
        float v = (p8[q] < KNN) ? s_mv[(rr * 8 + q) * KNN + p8[q]] : NEG_INF;
        if (v > best) { best = v; bs = q; }
      }
      outp[sel] = s_mi[(rr * 8 + bs) * KNN + p8[bs]];
      p8[bs] += 1;
    }
  }
}

// ---------------------------------------------------------------------------
// Kernel 2: edge features + conv(8->64) + BN1 + LeakyReLU + max over K
// 64 threads per point (one channel each), 4 points per 256-thread block.
// ---------------------------------------------------------------------------
__global__ __launch_bounds__(256) void edge_kernel(const float* __restrict__ pts,
                                                   const int* __restrict__ knn_idx,
                                                   const float* __restrict__ ew,
                                                   const float* __restrict__ g1,
                                                   const float* __restrict__ b1,
                                                   const float* __restrict__ m1,
                                                   const float* __restrict__ v1,
                                                   float* __restrict__ e_out) {
  const int tid = threadIdx.x;
  const int pt  = blockIdx.x * 4 + (tid >> 6);   // global flat point
  const int h   = tid & 63;
  const int b   = pt >> 12;

  const float ct0 = pts[pt * 4 + 0], ct1 = pts[pt * 4 + 1];
  const float ct2 = pts[pt * 4 + 2], ct3 = pts[pt * 4 + 3];
  const float w0 = ew[h * 8 + 0], w1 = ew[h * 8 + 1], w2 = ew[h * 8 + 2], w3 = ew[h * 8 + 3];
  const float w4 = ew[h * 8 + 4], w5 = ew[h * 8 + 5], w6 = ew[h * 8 + 6], w7 = ew[h * 8 + 7];
  const float sc = g1[h] * rsqrtf(v1[h] + 1e-5f);
  const float sh = b1[h] - m1[h] * sc;
  const float cbias = ct0 * w4 + ct1 * w5 + ct2 * w6 + ct3 * w7;

  float mx = NEG_INF;
  const int* ip = knn_idx + pt * KNN;
#pragma unroll 1
  for (int k = 0; k < KNN; ++k) {
    const int j = ip[k];
    const float* nb = pts + (b * NPTS + j) * 4;
    float sv = (nb[0] - ct0) * w0 + (nb[1] - ct1) * w1 +
               (nb[2] - ct2) * w2 + (nb[3] - ct3) * w3 + cbias;
    sv = sv * sc + sh;
    sv = (sv > 0.0f) ? sv : 0.2f * sv;
    mx = fmaxf(mx, sv);
  }
  e_out[pt * 64 + h] = mx;
}

// ---------------------------------------------------------------------------
// Kernel 3: quaternion linear as dense (N,64)@(64,128) via WMMA f32 16x16x4,
// + qbias + BN2 + exact GELU. Combined structured weight built in LDS.
// 8 waves/block, 16 points/wave -> 128 points/block.
// ---------------------------------------------------------------------------
__global__ __launch_bounds__(256) void quat_kernel(const float* __restrict__ e,
                                                   const float* __restrict__ wr,
                                                   const float* __restrict__ wi,
                                                   const float* __restrict__ wj,
                                                   const float* __restrict__ wk,
                                                   const float* __restrict__ qbias,
                                                   const float* __restrict__ gg2,
                                                   const float* __restrict__ bb2,
                                                   const float* __restrict__ mm2,
                                                   const float* __restrict__ vv2,
                                                   float* __restrict__ enc) {
  __shared__ float sW[64 * 128];   // 32 KB combined quaternion weight [cin][cout]
  const int tid = threadIdx.x;

  {
    const int msel[4][4] = {{0,1,2,3},{1,0,3,2},{2,3,0,1},{3,2,1,0}};
    const float sgn[4][4] = {{1.f,-1.f,-1.f,-1.f},{1.f,1.f,1.f,-1.f},
                             {1.f,-1.f,1.f,1.f},{1.f,1.f,-1.f,1.f}};
    for (int t = tid; t < 64 * 128; t += 256) {
      const int cin = t >> 7, cout = t & 127;
      const int ci = cin >> 4, ti = cin & 15;
      const int co = cout >> 5, u = cout & 31;
      const int ms = msel[co][ci];
      const float* mp = (ms == 0) ? wr : (ms == 1) ? wi : (ms == 2) ? wj : wk;
      sW[t] = sgn[co][ci] * mp[u * 16 + ti];
    }
  }
  __syncthreads();

  const int lane = tid & 31;
  const int wave = tid >> 5;
  const int n0   = blockIdx.x * 128 + wave * 16;  // global flat point
  const int mrow = lane & 15;
  const int kb   = (lane < 16) ? 0 : 2;
  const int hi   = (lane >> 4) << 3;

  // A fragments for all 16 K-slices (16x4 each), kept in registers
  v2f af[16];
#pragma unroll
  for (int kk = 0; kk < 16; ++kk) {
    af[kk].x = e[(n0 + mrow) * 64 + kk * 4 + kb];
    af[kk].y = e[(n0 + mrow) * 64 + kk * 4 + kb + 1];
  }

  for (int ct = 0; ct < 8; ++ct) {
    const int col = ct * 16 + mrow;
    v8f acc = {0.f, 0.f, 0.f, 0.f, 0.f, 0.f, 0.f, 0.f};
#pragma unroll
    for (int kk = 0; kk < 16; ++kk) {
      v2f bf;
      bf.x = sW[(kk * 4 + kb) * 128 + col];
      bf.y = sW[(kk * 4 + kb + 1) * 128 + col];
      acc = __builtin_amdgcn_wmma_f32_16x16x4_f32(false, af[kk], false, bf,
                                                  (short)0, acc, false, false);
    }
    const float sc = gg2[col] * rsqrtf(vv2[col] + 1e-5f);
    const float sh = bb2[col] - mm2[col] * sc;
    const float qb = qbias[col];
#pragma unroll
    for (int v = 0; v < 8; ++v) {
      const int row = v + hi;
      float x = acc[v] + qb;
      x = x * sc + sh;
      enc[(n0 + row) * 128 + col] = 0.5f * x * (1.0f + erff(x * 0.70710678118654752f));
    }
  }
}

// ---------------------------------------------------------------------------
// Kernel 4: merged = sum-of-squares over groups of 4, m = merged@merge_w.T,
// BN3 + GELU. In-place over the enc buffer (reads precede barrier).
// One 128-thread block per point.
// ---------------------------------------------------------------------------
__global__ __launch_bounds__(128) void merge_kernel(const float* __restrict__ mw,
                                                    const float* __restrict__ g3,
                                                    const float* __restrict__ b3,
                                                    const float* __restrict__ m3,
                                                    const float* __restrict__ v3,
                                                    float* __restrict__ encm) {
  __shared__ float mg[32];
  const int pt  = blockIdx.x;
  const int tid = threadIdx.x;
  float* row = encm + pt * 128;
  if (tid < 32) {
    const float a0 = row[tid * 4 + 0], a1 = row[tid * 4 + 1];
    const float a2 = row[tid * 4 + 2], a3 = row[tid * 4 + 3];
    mg[tid] = a0 * a0 + a1 * a1 + a2 * a2 + a3 * a3;
  }
  __syncthreads();
  float acc = 0.0f;
#pragma unroll
  for (int g = 0; g < 32; ++g) acc += mg[g] * mw[tid * 32 + g];
  const float sc = g3[tid] * rsqrtf(v3[tid] + 1e-5f);
  const float sh = b3[tid] - m3[tid] * sc;
  const float x = acc * sc + sh;
  row[tid] = 0.5f * x * (1.0f + erff(x * 0.70710678118654752f));
}

// ---------------------------------------------------------------------------
// Kernel 5: feats = [max over N, mean over N] per (b, channel)
// ---------------------------------------------------------------------------
__global__ __launch_bounds__(128) void pool_kernel(const float* __restrict__ m,
                                                   float* __restrict__ feats) {
  const int b = blockIdx.x, c = threadIdx.x;
  float mx = NEG_INF, sum = 0.0f;
#pragma unroll 1
  for (int n = 0; n < NPTS; ++n) {
    const float v = m[(b * NPTS + n) * 128 + c];
    mx = fmaxf(mx, v);
    sum += v;
  }
  feats[b * 256 + c] = mx;
  feats[b * 256 + 128 + c] = sum * (1.0f / (float)NPTS);
}

// ---------------------------------------------------------------------------
// Kernel 6: classifier (B,256)@(256,40)^T + bias
// ---------------------------------------------------------------------------
__global__ __launch_bounds__(64) void cls_kernel(const float* __restrict__ feats,
                                                 const float* __restrict__ cw,
                                                 const float* __restrict__ cb,
                                                 float* __restrict__ out) {
  const int b = blockIdx.x, j = threadIdx.x;
  if (j >= 40) return;
  float acc = cb[j];
#pragma unroll 1
  for (int q = 0; q < 256; ++q) acc += feats[b * 256 + q] * cw[j * 256 + q];
  out[b * 40 + j] = acc;
}

// ---------------------------------------------------------------------------
extern "C" void kernel_launch(void* const* d_in, const int* in_sizes, int n_in,
                              void* d_out, int out_size, void* d_ws, size_t ws_size,
                              hipStream_t stream) {
  const float* pts     = (const float*)d_in[0];
  const float* edge_w  = (const float*)d_in[1];
  const float* bn1_g   = (const float*)d_in[2];
  const float* bn1_b   = (const float*)d_in[3];
  const float* bn1_m   = (const float*)d_in[4];
  const float* bn1_v   = (const float*)d_in[5];
  const float* wr      = (const float*)d_in[6];
  const float* wi      = (const float*)d_in[7];
  const float* wj      = (const float*)d_in[8];
  const float* wk      = (const float*)d_in[9];
  const float* qbias   = (const float*)d_in[10];
  const float* bn2_g   = (const float*)d_in[11];
  const float* bn2_b   = (const float*)d_in[12];
  const float* bn2_m   = (const float*)d_in[13];
  const float* bn2_v   = (const float*)d_in[14];
  const float* merge_w = (const float*)d_in[15];
  const float* bn3_g   = (const float*)d_in[16];
  const float* bn3_b   = (const float*)d_in[17];
  const float* bn3_m   = (const float*)d_in[18];
  const float* bn3_v   = (const float*)d_in[19];
  const float* cls_w   = (const float*)d_in[20];
  const float* cls_b   = (const float*)d_in[21];

  char* ws = (char*)d_ws;
  float* xx    = (float*)(ws + 0);                       // 32768 f32
  int*   knn   = (int*)  (ws + 131072);                  // 32768*20 i32
  float* e_buf = (float*)(ws + 2752512);                 // 32768*64 f32
  float* enc   = (float*)(ws + 11141120);                // 32768*128 f32 (reused as m)
  float* feats = (float*)(ws + 27918336);                // 8*256 f32

  xx_kernel  <<<(NB * NPTS + 255) / 256, 256, 0, stream>>>(pts, xx);
  knn_kernel <<<NB * (NPTS / 16), 128, 0, stream>>>(pts, xx, knn);
  edge_kernel<<<(NB * NPTS) / 4, 256, 0, stream>>>(pts, knn, edge_w,
                                                   bn1_g, bn1_b, bn1_m, bn1_v, e_buf);
  quat_kernel<<<(NB * NPTS) / 128, 256, 0, stream>>>(e_buf, wr, wi, wj, wk, qbias,
                                                     bn2_g, bn2_b, bn2_m, bn2_v, enc);
  merge_kernel<<<NB * NPTS, 128, 0, stream>>>(merge_w, bn3_g, bn3_b, bn3_m, bn3_v, enc);
  pool_kernel<<<NB, 128, 0, stream>>>(enc, feats);
  cls_kernel <<<NB, 64, 0, stream>>>(feats, cls_w, cls_b, (float*)d_out);
}